// PhysicsGuidedCausalGCN_6966436954290
// MI455X (gfx1250) — compile-verified
//
#include <hip/hip_runtime.h>
#include <math.h>

typedef float v2f __attribute__((ext_vector_type(2)));
typedef float v8f __attribute__((ext_vector_type(8)));

namespace {
constexpr int kN    = 512;   // nodes
constexpr int kIn   = 64;    // input dim
constexpr int kH    = 128;   // hidden
constexpr int kB    = 2;     // batch
constexpr int kE    = 4096;  // directed edges
constexpr int kRows = kB * kN;
}

// ---------------------------------------------------------------------------
// Generic f32 WMMA GEMM: C = A(MxK) * B + bias   (one wave = one 16x16 tile)
// flags bit0: relu on output; bit1: B is stored (N x K) row-major (use B^T)
// ---------------------------------------------------------------------------
__global__ __launch_bounds__(32) void wmma_gemm_f32(
    const float* __restrict__ A, const float* __restrict__ B,
    const float* __restrict__ bias, float* __restrict__ C,
    int M, int N, int K, long sA, long sB, long sC, int flags)
{
  const int lane = threadIdx.x;
  const int half = lane >> 4;   // which 16-lane half
  const int l16  = lane & 15;
  A += (long)blockIdx.z * sA;
  B += (long)blockIdx.z * sB;
  C += (long)blockIdx.z * sC;
  const int rowA = blockIdx.y * 16 + l16;
  const int colB = blockIdx.x * 16 + l16;

  v8f acc = {};
  if (flags & 2) {
    const float* Brow = B + (long)colB * K;   // B^T: row colB of (N x K)
    for (int k = 0; k < K; k += 4) {
      const int ka = k + (half << 1);
      v2f a, b;
      a[0] = A[(long)rowA * K + ka];
      a[1] = A[(long)rowA * K + ka + 1];
      b[0] = Brow[ka];
      b[1] = Brow[ka + 1];
      acc = __builtin_amdgcn_wmma_f32_16x16x4_f32(false, a, false, b,
                                                  (short)0, acc, false, false);
    }
  } else {
    for (int k = 0; k < K; k += 4) {
      const int ka = k + (half << 1);
      v2f a, b;
      a[0] = A[(long)rowA * K + ka];
      a[1] = A[(long)rowA * K + ka + 1];
      b[0] = B[(long)ka * N + colB];
      b[1] = B[(long)(ka + 1) * N + colB];
      acc = __builtin_amdgcn_wmma_f32_16x16x4_f32(false, a, false, b,
                                                  (short)0, acc, false, false);
    }
  }

  const float bv = bias ? bias[colB] : 0.0f;
#pragma unroll
  for (int v = 0; v < 8; ++v) {
    const int rowC = blockIdx.y * 16 + v + (half << 3);
    float val = acc[v] + bv;
    if (flags & 1) val = fmaxf(val, 0.0f);
    C[(long)rowC * N + colB] = val;
  }
}

// ---------------------------------------------------------------------------
// Causal matrix: C = softmax(logits,row) * adj * dir; renorm rows (clip 1e-8).
// One block (256 threads) per row, 2 columns per thread (N=512).
// ---------------------------------------------------------------------------
__global__ __launch_bounds__(256) void causal_matrix_kernel(
    const float* __restrict__ logits, const float* __restrict__ adj,
    const float* __restrict__ dir, float* __restrict__ Cout)
{
  __shared__ float sh[256];
  const int r = blockIdx.x;
  const int t = threadIdx.x;
  const long base = (long)r * kN;
  const float l0 = logits[base + t];
  const float l1 = logits[base + t + 256];

  sh[t] = fmaxf(l0, l1); __syncthreads();
  for (int s = 128; s > 0; s >>= 1) { if (t < s) sh[t] = fmaxf(sh[t], sh[t + s]); __syncthreads(); }
  const float m = sh[0]; __syncthreads();

  const float e0 = expf(l0 - m), e1 = expf(l1 - m);
  sh[t] = e0 + e1; __syncthreads();
  for (int s = 128; s > 0; s >>= 1) { if (t < s) sh[t] += sh[t + s]; __syncthreads(); }
  const float se = sh[0]; __syncthreads();

  const float p0 = e0 / se * adj[base + t]       * dir[base + t];
  const float p1 = e1 / se * adj[base + t + 256] * dir[base + t + 256];
  sh[t] = p0 + p1; __syncthreads();
  for (int s = 128; s > 0; s >>= 1) { if (t < s) sh[t] += sh[t + s]; __syncthreads(); }
  const float inv = 1.0f / fmaxf(sh[0], 1e-8f);

  Cout[base + t]       = p0 * inv;
  Cout[base + t + 256] = p1 * inv;
}

// ---------------------------------------------------------------------------
// Attention: in-place over S (B,N,N):
//   p = softmax(S*rscale, row); a = p*C[row]; S = a / (sum(a) + 1e-8)
// ---------------------------------------------------------------------------
__global__ __launch_bounds__(256) void attn_softmax_kernel(
    float* __restrict__ S, const float* __restrict__ Cmat, float rscale)
{
  __shared__ float sh[256];
  const int r = blockIdx.x;
  const int b = blockIdx.y;
  const int t = threadIdx.x;
  float* srow = S + ((long)b * kN + r) * kN;
  const float* crow = Cmat + (long)r * kN;

  const float s0 = srow[t] * rscale;
  const float s1 = srow[t + 256] * rscale;
  sh[t] = fmaxf(s0, s1); __syncthreads();
  for (int s = 128; s > 0; s >>= 1) { if (t < s) sh[t] = fmaxf(sh[t], sh[t + s]); __syncthreads(); }
  const float m = sh[0]; __syncthreads();

  const float e0 = expf(s0 - m), e1 = expf(s1 - m);
  sh[t] = e0 + e1; __syncthreads();
  for (int s = 128; s > 0; s >>= 1) { if (t < s) sh[t] += sh[t + s]; __syncthreads(); }
  const float se = sh[0]; __syncthreads();

  const float a0 = e0 / se * crow[t];
  const float a1 = e1 / se * crow[t + 256];
  sh[t] = a0 + a1; __syncthreads();
  for (int s = 128; s > 0; s >>= 1) { if (t < s) sh[t] += sh[t + s]; __syncthreads(); }
  const float inv = 1.0f / (sh[0] + 1e-8f);

  srow[t]       = a0 * inv;
  srow[t + 256] = a1 * inv;
}

// ---------------------------------------------------------------------------
// GCN helpers
// ---------------------------------------------------------------------------
__global__ void fill_kernel(float* __restrict__ p, float v, int n) {
  const int i = blockIdx.x * blockDim.x + threadIdx.x;
  if (i < n) p[i] = v;
}

__global__ void deg_accum_kernel(const int* __restrict__ dst, float* __restrict__ deg, int E) {
  const int e = blockIdx.x * blockDim.x + threadIdx.x;
  if (e < E) atomicAdd(&deg[dst[e]], 1.0f);
}

__global__ void dinv_kernel(const float* __restrict__ deg, float* __restrict__ dinv, int n) {
  const int i = blockIdx.x * blockDim.x + threadIdx.x;
  if (i < n) dinv[i] = 1.0f / sqrtf(deg[i]);
}

// grid: (E + N, B); threads: H. e >= E are the self-loop entries.
__global__ __launch_bounds__(128) void gcn_scatter_kernel(
    const float* __restrict__ hW, const int* __restrict__ src,
    const int* __restrict__ dst, const float* __restrict__ dinv,
    float* __restrict__ out, int E)
{
  const int e = blockIdx.x;
  const int b = blockIdx.y;
  const int t = threadIdx.x;
  int s, d;
  if (e < E) { s = src[e]; d = dst[e]; } else { s = d = e - E; }
  const float nrm = dinv[s] * dinv[d];
  const float msg = hW[((long)b * kN + s) * kH + t] * nrm;
  atomicAdd(&out[((long)b * kN + d) * kH + t], msg);
}

__global__ void bias_relu_kernel(float* __restrict__ x, const float* __restrict__ bias, int n) {
  const int i = blockIdx.x * blockDim.x + threadIdx.x;
  if (i < n) x[i] = fmaxf(x[i] + bias[i & (kH - 1)], 0.0f);
}

// Build [h_causal | h_g] rows of width 2H. grid: rows, threads: 256
__global__ __launch_bounds__(256) void concat_kernel(
    const float* __restrict__ hc, const float* __restrict__ hg, float* __restrict__ out)
{
  const int r = blockIdx.x;
  const int t = threadIdx.x;
  out[(long)r * (2 * kH) + t] = (t < kH) ? hc[(long)r * kH + t]
                                         : hg[(long)r * kH + (t - kH)];
}

// LayerNorm over last dim (H=128). grid: rows, threads: 128
__global__ __launch_bounds__(128) void layernorm_kernel(
    const float* __restrict__ X, const float* __restrict__ g,
    const float* __restrict__ b, float* __restrict__ out)
{
  __shared__ float sh[128];
  const int r = blockIdx.x;
  const int t = threadIdx.x;
  const float v = X[(long)r * kH + t];
  sh[t] = v; __syncthreads();
  for (int s = 64; s > 0; s >>= 1) { if (t < s) sh[t] += sh[t + s]; __syncthreads(); }
  const float mu = sh[0] / kH; __syncthreads();
  const float d = v - mu;
  sh[t] = d * d; __syncthreads();
  for (int s = 64; s > 0; s >>= 1) { if (t < s) sh[t] += sh[t + s]; __syncthreads(); }
  const float var = sh[0] / kH;
  out[(long)r * kH + t] = d / sqrtf(var + 1e-5f) * g[t] + b[t];
}

// ---------------------------------------------------------------------------
// Orchestration
// ---------------------------------------------------------------------------
extern "C" void kernel_launch(void* const* d_in, const int* in_sizes, int n_in,
                              void* d_out, int out_size, void* d_ws, size_t ws_size,
                              hipStream_t stream) {
  (void)in_sizes; (void)n_in; (void)out_size; (void)ws_size;

  const float* x       = (const float*)d_in[0];
  const int*   eidx    = (const int*)d_in[1];          // (2, 4096)
  const float* adj     = (const float*)d_in[2];
  const float* dir     = (const float*)d_in[3];
  const float* logits  = (const float*)d_in[4];
  const float* W_in    = (const float*)d_in[5];
  const float* b_in    = (const float*)d_in[6];
  const float* Wq      = (const float*)d_in[7];
  const float* bq      = (const float*)d_in[8];
  const float* Wk      = (const float*)d_in[9];
  const float* bk      = (const float*)d_in[10];
  const float* Wv      = (const float*)d_in[11];
  const float* bv      = (const float*)d_in[12];
  const float* gcn_W   = (const float*)d_in[13];       // (2,128,128)
  const float* gcn_b   = (const float*)d_in[14];       // (2,128)
  const float* W_fuse  = (const float*)d_in[15];       // (256,128)
  const float* b_fuse  = (const float*)d_in[16];
  const float* ln_g    = (const float*)d_in[17];
  const float* ln_b    = (const float*)d_in[18];

  float* out  = (float*)d_out;                         // (B,N,H) = 131072
  float* Cout = out + (long)kB * kN * kH;              // (N,N)   = 262144

  // workspace layout (floats)
  float* ws    = (float*)d_ws;
  float* h     = ws;                                   // 131072
  float* Qb    = ws + 131072;                          // 131072
  float* Kb    = ws + 262144;                          // 131072
  float* Vb    = ws + 393216;                          // 131072
  float* Sb    = ws + 524288;                          // 524288 (B,N,N)
  float* hc    = ws + 1048576;                         // 131072 (h_causal)
  float* deg   = ws + 1179648;                         // 512
  float* dinv  = ws + 1180160;                         // 512
  // reuse after attention is done:
  float* hgA   = Qb;                                   // gcn features ping
  float* hW    = Kb;                                   // gcn matmul temp
  float* hgB   = Vb;                                   // gcn features pong
  float* hcat  = Sb;                                   // 1024 x 256 concat
  float* hcomb = Sb + 262144;                          // 131072

  const int* e_src = eidx;
  const int* e_dst = eidx + kE;

  // --- degree / normalization (needed by GCN scatter later) ---
  fill_kernel<<<2, 256, 0, stream>>>(deg, 1.0f, kN);   // self loops
  deg_accum_kernel<<<(kE + 255) / 256, 256, 0, stream>>>(e_dst, deg, kE);
  dinv_kernel<<<2, 256, 0, stream>>>(deg, dinv, kN);

  // --- causal matrix C (second output; also used by attention) ---
  causal_matrix_kernel<<<kN, 256, 0, stream>>>(logits, adj, dir, Cout);

  // --- h = relu(x @ W_in + b_in)   (1024 x 128, K=64) ---
  wmma_gemm_f32<<<dim3(kH / 16, kRows / 16, 1), 32, 0, stream>>>(
      x, W_in, b_in, h, kRows, kH, kIn, 0, 0, 0, /*relu*/1);

  // --- Q, K, V projections (1024 x 128, K=128) ---
  wmma_gemm_f32<<<dim3(kH / 16, kRows / 16, 1), 32, 0, stream>>>(
      h, Wq, bq, Qb, kRows, kH, kH, 0, 0, 0, 0);
  wmma_gemm_f32<<<dim3(kH / 16, kRows / 16, 1), 32, 0, stream>>>(
      h, Wk, bk, Kb, kRows, kH, kH, 0, 0, 0, 0);
  wmma_gemm_f32<<<dim3(kH / 16, kRows / 16, 1), 32, 0, stream>>>(
      h, Wv, bv, Vb, kRows, kH, kH, 0, 0, 0, 0);

  // --- S = Q @ K^T per batch (512 x 512, K=128) ---
  wmma_gemm_f32<<<dim3(kN / 16, kN / 16, kB), 32, 0, stream>>>(
      Qb, Kb, nullptr, Sb, kN, kN, kH,
      (long)kN * kH, (long)kN * kH, (long)kN * kN, /*Bt*/2);

  // --- masked softmax + renorm (in place on S) ---
  attn_softmax_kernel<<<dim3(kN, kB), 256, 0, stream>>>(Sb, Cout, 1.0f / sqrtf((float)kH));

  // --- h_causal = attn @ V per batch (512 x 128, K=512) ---
  wmma_gemm_f32<<<dim3(kH / 16, kN / 16, kB), 32, 0, stream>>>(
      Sb, Vb, nullptr, hc, kN, kH, kN,
      (long)kN * kN, (long)kN * kH, (long)kN * kH, 0);

  // --- GCN layer 0: hW = h @ W0 ; scatter ; bias+relu -> hgA ---
  wmma_gemm_f32<<<dim3(kH / 16, kRows / 16, 1), 32, 0, stream>>>(
      h, gcn_W, nullptr, hW, kRows, kH, kH, 0, 0, 0, 0);
  fill_kernel<<<(kRows * kH + 255) / 256, 256, 0, stream>>>(hgA, 0.0f, kRows * kH);
  gcn_scatter_kernel<<<dim3(kE + kN, kB), kH, 0, stream>>>(hW, e_src, e_dst, dinv, hgA, kE);
  bias_relu_kernel<<<(kRows * kH + 255) / 256, 256, 0, stream>>>(hgA, gcn_b, kRows * kH);

  // --- GCN layer 1: hW = hgA @ W1 ; scatter ; bias+relu -> hgB ---
  wmma_gemm_f32<<<dim3(kH / 16, kRows / 16, 1), 32, 0, stream>>>(
      hgA, gcn_W + kH * kH, nullptr, hW, kRows, kH, kH, 0, 0, 0, 0);
  fill_kernel<<<(kRows * kH + 255) / 256, 256, 0, stream>>>(hgB, 0.0f, kRows * kH);
  gcn_scatter_kernel<<<dim3(kE + kN, kB), kH, 0, stream>>>(hW, e_src, e_dst, dinv, hgB, kE);
  bias_relu_kernel<<<(kRows * kH + 255) / 256, 256, 0, stream>>>(hgB, gcn_b + kH, kRows * kH);

  // --- fuse: [h_causal | h_g] @ W_fuse + b_fuse (1024 x 128, K=256) ---
  concat_kernel<<<kRows, 256, 0, stream>>>(hc, hgB, hcat);
  wmma_gemm_f32<<<dim3(kH / 16, kRows / 16, 1), 32, 0, stream>>>(
      hcat, W_fuse, b_fuse, hcomb, kRows, kH, 2 * kH, 0, 0, 0, 0);

  // --- layernorm -> output 0 ---
  layernorm_kernel<<<kRows, kH, 0, stream>>>(hcomb, ln_g, ln_b, out);
}